// RecyclingEmbedder_7215545057299
// MI455X (gfx1250) — compile-verified
//
#include <hip/hip_runtime.h>
#include <hip/hip_bf16.h>
#include <math.h>

typedef __attribute__((ext_vector_type(2))) float v2f;
typedef __attribute__((ext_vector_type(8))) float v8f;

#define NRES  768
#define PAIRC 128
#define MSAC  256
#define NFEAT 88      // 15 distogram bins + 66 rel_pos + 1 entity_same + 6 rel_chain
#define KSTEPS 22     // 88 / 4
#define NPAIRS 44     // feature rows stored as pairs
#define PSTR  144     // pair-row stride in float2 units (288 dwords == 32 mod 64 banks)

static __device__ __forceinline__ int iclamp(int v, int lo, int hi) {
  return v < lo ? lo : (v > hi ? hi : v);
}

// ---------------- kernel 0: pseudo-beta gather -> ws ----------------
__global__ void pb_kernel(const int* __restrict__ aatype,
                          const float* __restrict__ prev_pos,
                          float4* __restrict__ pb) {
  int i = blockIdx.x * blockDim.x + threadIdx.x;
  if (i >= NRES) return;
  const float* base = prev_pos + (size_t)i * (37 * 3);
  int atom = (aatype[i] == 7) ? 1 : 3;  // CA for GLY else CB
  float4 r;
  r.x = base[atom * 3 + 0];
  r.y = base[atom * 3 + 1];
  r.z = base[atom * 3 + 2];
  r.w = 0.0f;
  pb[i] = r;
}

// ---------------- kernel 1: msa layernorm (one block per row) ----------------
__global__ __launch_bounds__(256) void msa_kernel(const float* __restrict__ x,
                                                  const float* __restrict__ g,
                                                  const float* __restrict__ b,
                                                  float* __restrict__ out) {
  __shared__ float red[16];
  int row = blockIdx.x;
  int t = threadIdx.x;
  float v = x[(size_t)row * MSAC + t];
  float s = v, s2 = v * v;
#pragma unroll
  for (int o = 16; o > 0; o >>= 1) { s += __shfl_xor(s, o); s2 += __shfl_xor(s2, o); }
  int wid = t >> 5;
  if ((t & 31) == 0) { red[wid] = s; red[8 + wid] = s2; }
  __syncthreads();
  float ts = 0.0f, ts2 = 0.0f;
#pragma unroll
  for (int w = 0; w < 8; w++) { ts += red[w]; ts2 += red[8 + w]; }
  float mu = ts * (1.0f / MSAC);
  float var = ts2 * (1.0f / MSAC) - mu * mu;
  float rs = 1.0f / sqrtf(var + 1e-5f);
  out[(size_t)row * MSAC + t] = (v - mu) * rs * g[t] + b[t];
}

// ---------------- kernel 2: pair (WMMA one-hot einsums + fused layernorm) ----------------
__global__ __launch_bounds__(256) void pair_kernel(
    const int* __restrict__ ri, const int* __restrict__ asym,
    const int* __restrict__ ent, const int* __restrict__ sym,
    const float4* __restrict__ pb, const float* __restrict__ prev_pair,
    const float* __restrict__ w_dgram, const float* __restrict__ b_dgram,
    const float* __restrict__ g_pair, const float* __restrict__ bt_pair,
    const float* __restrict__ w_rel, const float* __restrict__ b_rel,
    float* __restrict__ pair_out) {
  // Weight table stored as feature-PAIRS: WP[p][c] = {W[2p][c], W[2p+1][c]}
  // so a B fragment is one ds_load_b64.  Pair-row stride 144*float2 keeps the
  // two lane-halves (p = 2k vs 2k+1) on disjoint bank halves.
  __shared__ float WP[NPAIRS * PSTR * 2];
  __shared__ float gl[PAIRC], bias2[PAIRC];

  int tid = threadIdx.x;
  for (int idx = tid; idx < NPAIRS * PAIRC; idx += 256) {
    int p = idx >> 7, c = idx & 127;
    int f0 = 2 * p, f1 = 2 * p + 1;
    float v0 = (f0 < 15) ? w_dgram[f0 * PAIRC + c] : w_rel[(f0 - 15) * PAIRC + c];
    float v1 = (f1 < 15) ? w_dgram[f1 * PAIRC + c] : w_rel[(f1 - 15) * PAIRC + c];
    WP[(p * PSTR + c) * 2 + 0] = v0;
    WP[(p * PSTR + c) * 2 + 1] = v1;
  }
  if (tid < PAIRC) {
    gl[tid] = g_pair[tid];
    bias2[tid] = bt_pair[tid] + b_dgram[tid] + b_rel[tid];
  }
  __syncthreads();

  int wave = tid >> 5, lane = tid & 31;
  int strip = blockIdx.x * 8 + wave;   // one wave = one (i, 16-wide j strip)
  int i = strip / 48;
  int j0 = (strip % 48) * 16;
  int m = lane & 15, half = lane >> 4;
  int j = j0 + m;

  // ---- per-row feature codes (lanes 0-15 and 16-31 duplicate rows 0..15) ----
  float4 pi = pb[i], pj = pb[j];
  float dx = pi.x - pj.x, dy = pi.y - pj.y, dz = pi.z - pj.z;
  // force mul/add (no fma) so bin-boundary semantics match the reference exactly
  float d2 = __fadd_rn(__fadd_rn(__fmul_rn(dx, dx), __fmul_rn(dy, dy)), __fmul_rn(dz, dz));
  int cnt = 0;
  bool eq = false;
#pragma unroll
  for (int bb = 0; bb < 15; bb++) {
    float lo = 3.25f + 1.25f * (float)bb;   // exact in fp32
    lo = __fmul_rn(lo, lo);                 // single rounding, same as jnp square
    cnt += (d2 > lo) ? 1 : 0;
    eq = eq || (d2 == lo);
  }
  bool valid = (cnt > 0) && !eq && (cnt < 15 || d2 < 1e8f);

  int c_off = (asym[i] == asym[j]) ? (15 + iclamp(ri[i] - ri[j] + 32, 0, 64)) : 80;
  bool es = (ent[i] == ent[j]);
  int c_chain = 82 + (es ? iclamp(sym[i] - sym[j] + 2, 0, 4) : 5);

  // ---- pack hot feature indices into a 88-bit mask (lo: f0..63, hi: f64..87) ----
  unsigned long long lomask = 0ull;
  unsigned int himask = 0u;
  if (valid) lomask |= (1ull << (cnt - 1));          // distogram bin 0..14
  if (c_off < 64) lomask |= (1ull << c_off); else himask |= (1u << (c_off - 64));
  if (es) himask |= (1u << (81 - 64));               // entity_same feature
  himask |= (1u << (c_chain - 64));                  // rel_chain 82..87

  // ---- one-hot A fragments (A 16x4 f32 layout: v0 <- K0/K2, v1 <- K1/K3) ----
  v2f A[KSTEPS];
#pragma unroll
  for (int k = 0; k < KSTEPS; k++) {
    int f0 = 4 * k + 2 * half;
    unsigned int bits;
    if (k < 16) bits = (unsigned int)(lomask >> f0);   // f0<=62, f1<=63: compile-time split
    else        bits = (himask >> (f0 - 64));
    A[k].x = (bits & 1u) ? 1.0f : 0.0f;
    A[k].y = (bits & 2u) ? 1.0f : 0.0f;
  }

  // ---- layernorm stats for the 16 prev_pair rows of this strip ----
  const size_t rowbase = ((size_t)i * NRES + j0) * PAIRC;
  float mu_keep = 0.0f, rs_keep = 0.0f;
  for (int mm = 0; mm < 16; mm++) {
    const float4 xv = *(const float4*)(prev_pair + rowbase + (size_t)mm * PAIRC + lane * 4);
    float s = xv.x + xv.y + xv.z + xv.w;
    float s2 = xv.x * xv.x + xv.y * xv.y + xv.z * xv.z + xv.w * xv.w;
#pragma unroll
    for (int o = 16; o > 0; o >>= 1) { s += __shfl_xor(s, o); s2 += __shfl_xor(s2, o); }
    if (m == mm) {
      float mu = s * (1.0f / PAIRC);
      float var = s2 * (1.0f / PAIRC) - mu * mu;
      mu_keep = mu;
      rs_keep = 1.0f / sqrtf(var + 1e-5f);
    }
  }
  // lane l holds stats for row (l & 15); gather the 8 rows this lane owns in D layout
  float muv[8], rsv[8];
#pragma unroll
  for (int v = 0; v < 8; v++) {
    int mrow = v + 8 * half;  // D layout: VGPR v <- row v (lanes 0-15) / v+8 (lanes 16-31)
    muv[v] = __shfl(mu_keep, mrow);
    rsv[v] = __shfl(rs_keep, mrow);
  }

  // ---- 8 channel tiles of 16: K=88 via 22 x V_WMMA_F32_16X16X4_F32 ----
  const size_t myrow = rowbase + (size_t)(8 * half) * PAIRC;
  for (int t8 = 0; t8 < 8; t8++) {
    int c = t8 * 16 + m;
    // preload all B fragments for this tile: 22 x ds_load_b64, one wait, then WMMA chain
    v2f Bf[KSTEPS];
#pragma unroll
    for (int k = 0; k < KSTEPS; k++) {
      int p = 2 * k + half;
      Bf[k] = *(const v2f*)(WP + (size_t)(p * PSTR + c) * 2);
    }
    v8f acc = {0.f, 0.f, 0.f, 0.f, 0.f, 0.f, 0.f, 0.f};
#pragma unroll
    for (int k = 0; k < KSTEPS; k++) {
      acc = __builtin_amdgcn_wmma_f32_16x16x4_f32(false, A[k], false, Bf[k], (short)0, acc,
                                                  false, false);
    }
    // fused epilogue: acc + (pp - mu)*rs*g + (bt + b_dgram + b_rel)
    float G = gl[c], BB = bias2[c];
    const float* ppp = prev_pair + myrow + c;
    float* outp = pair_out + myrow + c;
#pragma unroll
    for (int v = 0; v < 8; v++) {
      float rg = rsv[v] * G;
      float bb = fmaf(-muv[v], rg, BB);
      float ppv = ppp[(size_t)v * PAIRC];
      float r = fmaf(ppv, rg, bb) + acc[v];
      __builtin_nontemporal_store(r, outp + (size_t)v * PAIRC);  // streamed, never re-read
    }
  }
}

// ---------------- launch ----------------
extern "C" void kernel_launch(void* const* d_in, const int* in_sizes, int n_in,
                              void* d_out, int out_size, void* d_ws, size_t ws_size,
                              hipStream_t stream) {
  (void)in_sizes; (void)n_in; (void)out_size; (void)ws_size;
  const int*   aatype    = (const int*)d_in[0];
  const int*   ri        = (const int*)d_in[1];
  const int*   asym      = (const int*)d_in[2];
  const int*   ent       = (const int*)d_in[3];
  const int*   sym       = (const int*)d_in[4];
  const float* prev_pos  = (const float*)d_in[5];
  const float* prev_pair = (const float*)d_in[6];
  const float* prev_msa  = (const float*)d_in[7];
  const float* w_dgram   = (const float*)d_in[8];
  const float* b_dgram   = (const float*)d_in[9];
  const float* g_pair    = (const float*)d_in[10];
  const float* bt_pair   = (const float*)d_in[11];
  const float* g_msa     = (const float*)d_in[12];
  const float* bt_msa    = (const float*)d_in[13];
  const float* w_rel     = (const float*)d_in[14];
  const float* b_rel     = (const float*)d_in[15];

  float* out = (float*)d_out;
  float* msa_out = out;                    // first output: [768,256]
  float* pair_out = out + NRES * MSAC;     // second output: [768,768,128]
  float4* pb = (float4*)d_ws;

  pb_kernel<<<3, 256, 0, stream>>>(aatype, prev_pos, pb);
  msa_kernel<<<NRES, 256, 0, stream>>>(prev_msa, g_msa, bt_msa, msa_out);
  pair_kernel<<<(NRES * 48) / 8, 256, 0, stream>>>(ri, asym, ent, sym, pb, prev_pair,
                                                   w_dgram, b_dgram, g_pair, bt_pair,
                                                   w_rel, b_rel, pair_out);
}